// NodeGINConvNet_9036611191475
// MI455X (gfx1250) — compile-verified
//
#include <hip/hip_runtime.h>

// ---------------------------------------------------------------------------
// GIN network on gfx1250 (MI455X).
//  - Edge aggregation: global_atomic_add_f32, L2-resident (feature maps are
//    16-64 MB << 192 MB L2). Edge indices read via scalar loads (wave-uniform).
//  - MLPs + final FC: V_WMMA_F32_16X16X4_F32 (pure f32, matches reference).
//    Weights pre-packed into WMMA-B-pair layout so every operand fetch is a
//    single b64 load.
//  - BatchNorm stats: LDS ds_add_f32 block reduction, then global atomics.
// ---------------------------------------------------------------------------

#define NN   131072
#define EE   4194304
#define FIN  128
#define DIM  32
#define OUTD 1024
#define BN_EPS 1e-5f

typedef float v2f __attribute__((ext_vector_type(2)));
typedef float v8f __attribute__((ext_vector_type(8)));

__device__ __forceinline__ v8f wmma4(v2f a, v2f b, v8f c) {
  // D(16x16,f32) += A(16x4,f32) x B(4x16,f32)
  return __builtin_amdgcn_wmma_f32_16x16x4_f32(
      /*neg_a=*/false, a, /*neg_b=*/false, b,
      /*c_mod=*/(short)0, c, /*reuse_a=*/false, /*reuse_b=*/false);
}

// ------------------------------- utility -----------------------------------

__global__ void k_zero(float* __restrict__ p, int n) {
  int i = blockIdx.x * blockDim.x + threadIdx.x;
  if (i < n) p[i] = 0.0f;
}

__global__ void k_copy4(const float4* __restrict__ src, float4* __restrict__ dst, int n4) {
  int i = blockIdx.x * blockDim.x + threadIdx.x;
  if (i < n4) dst[i] = src[i];
}

// Pack row-major weight w[K][Ncols] into WMMA-B-pair layout:
// pw[p*Ncols + n] = { w[2p][n], w[2p+1][n] }   (p = 0..K/2-1)
__global__ void k_packB(const float* __restrict__ w, v2f* __restrict__ pw,
                        int K, int Ncols) {
  int i = blockIdx.x * blockDim.x + threadIdx.x;
  int total = (K / 2) * Ncols;
  if (i >= total) return;
  int p = i / Ncols;
  int n = i - p * Ncols;
  v2f v;
  v.x = w[(2 * p) * Ncols + n];
  v.y = w[(2 * p + 1) * Ncols + n];
  pw[i] = v;
}

// --------------------------- edge aggregation ------------------------------
// h[dst] += x[src].  h pre-initialized to x (GIN eps=0: h = x + sum agg).
// One wave per edge; edge index forced wave-uniform -> scalar loads.

__global__ void __launch_bounds__(256) k_scatter128(const float* __restrict__ x,
                                                    const int* __restrict__ srcI,
                                                    const int* __restrict__ dstI,
                                                    float* __restrict__ h) {
  const int lane = threadIdx.x & 31;
  const unsigned wv = (unsigned)__builtin_amdgcn_readfirstlane(threadIdx.x >> 5);
  const unsigned e = blockIdx.x * 8u + wv;           // EE/8 blocks
  const size_t s = (size_t)srcI[e];
  const size_t d = (size_t)dstI[e];
  float4 v = ((const float4*)(x + s * FIN))[lane];
  float* p = h + d * FIN + lane * 4;
  atomicAdd(p + 0, v.x);
  atomicAdd(p + 1, v.y);
  atomicAdd(p + 2, v.z);
  atomicAdd(p + 3, v.w);
}

__global__ void __launch_bounds__(256) k_scatter32(const float* __restrict__ x,
                                                   const int* __restrict__ srcI,
                                                   const int* __restrict__ dstI,
                                                   float* __restrict__ h) {
  const int c = threadIdx.x & 31;
  const unsigned wv = (unsigned)__builtin_amdgcn_readfirstlane(threadIdx.x >> 5);
  const unsigned e = blockIdx.x * 8u + wv;           // EE/8 blocks
  const size_t s = (size_t)srcI[e];
  const size_t d = (size_t)dstI[e];
  atomicAdd(h + d * DIM + c, x[s * DIM + c]);
}

// ------------------------------ fused MLP ----------------------------------
// out = relu(relu(hin @ wA + bA) @ wB + bB); per-channel sum/sumsq for BN
// reduced in LDS, then flushed to stats[0..31]=sum, stats[32..63]=sumsq.
// One wave handles 16 rows; 8 waves / block. Weights are pre-packed (v2f).

template <int FI>
__global__ void __launch_bounds__(256) k_mlp(const float* __restrict__ hin,
                                             const v2f* __restrict__ pwA,
                                             const float* __restrict__ bA,
                                             const v2f* __restrict__ pwB,
                                             const float* __restrict__ bB,
                                             float* __restrict__ hout,
                                             float* __restrict__ stats) {
  __shared__ float lds[8][16 * DIM];
  __shared__ float bstat[64];
  const int wave = threadIdx.x >> 5;
  const int lane = threadIdx.x & 31;
  const int half = lane >> 4;          // 0: lanes 0-15, 1: lanes 16-31
  const int l15  = lane & 15;
  const int row0 = (blockIdx.x * 8 + wave) * 16;
  const size_t arow = (size_t)(row0 + l15);

  if (threadIdx.x < 64) bstat[threadIdx.x] = 0.0f;

  v8f c0 = {0, 0, 0, 0, 0, 0, 0, 0};
  v8f c1 = {0, 0, 0, 0, 0, 0, 0, 0};

  // GEMM1: (16 x FI) @ (FI x 32), two 16x16 output tiles (n0 = 0, 16)
  const v2f* Arow = (const v2f*)(hin + arow * FI);
  for (int kc = 0; kc < FI / 4; ++kc) {
    const int p = 2 * kc + half;       // K-pair index for this half-wave
    v2f a  = Arow[p];
    v2f b0 = pwA[p * DIM + l15];
    v2f b1 = pwA[p * DIM + 16 + l15];
    c0 = wmma4(a, b0, c0);
    c1 = wmma4(a, b1, c1);
  }

  // bias + relu, stage 16x32 tile row-major into LDS (re-layout C/D -> A)
  float* S = lds[wave];
  #pragma unroll
  for (int r = 0; r < 8; ++r) {
    const int m = r + half * 8;
    float x0 = c0[r] + bA[l15];
    float x1 = c1[r] + bA[16 + l15];
    S[m * DIM + l15]      = fmaxf(x0, 0.0f);
    S[m * DIM + 16 + l15] = fmaxf(x1, 0.0f);
  }
  __syncthreads();

  // GEMM2: (16 x 32) @ (32 x 32)
  v8f d0 = {0, 0, 0, 0, 0, 0, 0, 0};
  v8f d1 = {0, 0, 0, 0, 0, 0, 0, 0};
  const v2f* Sa = (const v2f*)(S + l15 * DIM);
  #pragma unroll
  for (int kc = 0; kc < DIM / 4; ++kc) {
    const int p = 2 * kc + half;
    v2f a  = Sa[p];
    v2f b0 = pwB[p * DIM + l15];
    v2f b1 = pwB[p * DIM + 16 + l15];
    d0 = wmma4(a, b0, d0);
    d1 = wmma4(a, b1, d1);
  }

  // bias + relu + store + BN statistics (LDS reduction first)
  float s0 = 0.f, q0 = 0.f, s1 = 0.f, q1 = 0.f;
  #pragma unroll
  for (int r = 0; r < 8; ++r) {
    const int m = r + half * 8;
    float x0 = fmaxf(d0[r] + bB[l15], 0.0f);
    float x1 = fmaxf(d1[r] + bB[16 + l15], 0.0f);
    hout[(size_t)(row0 + m) * DIM + l15]      = x0;
    hout[(size_t)(row0 + m) * DIM + 16 + l15] = x1;
    s0 += x0; q0 += x0 * x0;
    s1 += x1; q1 += x1 * x1;
  }
  atomicAdd(&bstat[l15],           s0);   // ds_add_f32
  atomicAdd(&bstat[32 + l15],      q0);
  atomicAdd(&bstat[16 + l15],      s1);
  atomicAdd(&bstat[32 + 16 + l15], q1);
  __syncthreads();
  if (threadIdx.x < 64) atomicAdd(&stats[threadIdx.x], bstat[threadIdx.x]);
}

// ---------------------------- BatchNorm ------------------------------------

__global__ void k_bn_fin(const float* __restrict__ stats,
                         const float* __restrict__ gamma,
                         const float* __restrict__ beta,
                         float* __restrict__ ss) {
  int c = threadIdx.x;  // 32 threads
  float inv_n = 1.0f / (float)NN;
  float mean = stats[c] * inv_n;
  float var  = stats[32 + c] * inv_n - mean * mean;
  float scale = gamma[c] * rsqrtf(var + BN_EPS);
  ss[c]      = scale;
  ss[32 + c] = beta[c] - mean * scale;
}

// BN apply in-place, plus duplicate into next layer's aggregation buffer
// (fuses the h = x copy of the next GIN layer).
__global__ void k_bn_apply_dup(float* __restrict__ h, float* __restrict__ dup,
                               const float* __restrict__ ss) {
  unsigned i = blockIdx.x * blockDim.x + threadIdx.x;  // NN*DIM/4 threads
  float4 v = ((const float4*)h)[i];
  int c = (i * 4) & (DIM - 1);
  v.x = v.x * ss[c + 0] + ss[32 + c + 0];
  v.y = v.y * ss[c + 1] + ss[32 + c + 1];
  v.z = v.z * ss[c + 2] + ss[32 + c + 2];
  v.w = v.w * ss[c + 3] + ss[32 + c + 3];
  ((float4*)h)[i] = v;
  ((float4*)dup)[i] = v;
}

__global__ void k_bn_apply(float* __restrict__ h, const float* __restrict__ ss) {
  unsigned i = blockIdx.x * blockDim.x + threadIdx.x;  // NN*DIM/4 threads
  float4 v = ((const float4*)h)[i];
  int c = (i * 4) & (DIM - 1);
  v.x = v.x * ss[c + 0] + ss[32 + c + 0];
  v.y = v.y * ss[c + 1] + ss[32 + c + 1];
  v.z = v.z * ss[c + 2] + ss[32 + c + 2];
  v.w = v.w * ss[c + 3] + ss[32 + c + 3];
  ((float4*)h)[i] = v;
}

// ------------------------------ final FC -----------------------------------
// out = relu(h @ fc_w + fc_b); h: N x 32, fc_w packed (16 x 1024 v2f pairs).
// One wave computes a 16-row x 128-col tile: 8 accumulators, 8x8 WMMAs.

__global__ void __launch_bounds__(256) k_fc(const float* __restrict__ h,
                                            const v2f* __restrict__ pw,
                                            const float* __restrict__ bias,
                                            float* __restrict__ out) {
  const int wave = threadIdx.x >> 5;
  const int lane = threadIdx.x & 31;
  const int half = lane >> 4;
  const int l15  = lane & 15;
  const unsigned gw = blockIdx.x * 8 + wave;     // 65536 wave tiles
  const int row0 = (int)(gw >> 3) * 16;          // 8192 row tiles
  const int col0 = (int)(gw & 7) * 128;          // 8 col tiles of 128
  const size_t arow = (size_t)(row0 + l15);

  v8f acc[8];
  #pragma unroll
  for (int t = 0; t < 8; ++t) acc[t] = (v8f){0, 0, 0, 0, 0, 0, 0, 0};

  const v2f* Arow = (const v2f*)(h + arow * DIM);
  #pragma unroll
  for (int kc = 0; kc < DIM / 4; ++kc) {
    const int p = 2 * kc + half;
    v2f a = Arow[p];
    #pragma unroll
    for (int t = 0; t < 8; ++t) {
      const int n = col0 + t * 16 + l15;
      v2f b = pw[(size_t)p * OUTD + n];
      acc[t] = wmma4(a, b, acc[t]);
    }
  }

  #pragma unroll
  for (int t = 0; t < 8; ++t) {
    const int n = col0 + t * 16 + l15;
    const float bv = bias[n];
    #pragma unroll
    for (int r = 0; r < 8; ++r) {
      const int m = r + half * 8;
      out[(size_t)(row0 + m) * OUTD + n] = fmaxf(acc[t][r] + bv, 0.0f);
    }
  }
}

// ------------------------------- launcher ----------------------------------

extern "C" void kernel_launch(void* const* d_in, const int* in_sizes, int n_in,
                              void* d_out, int out_size, void* d_ws, size_t ws_size,
                              hipStream_t stream) {
  const float* node = (const float*)d_in[0];
  const int*   ei   = (const int*)d_in[1];     // [2, E]: src row then dst row
  const float* w1a  = (const float*)d_in[2];
  const float* b1a  = (const float*)d_in[3];
  const float* w1b  = (const float*)d_in[4];
  const float* b1b  = (const float*)d_in[5];
  const float* g1   = (const float*)d_in[6];
  const float* bt1  = (const float*)d_in[7];
  const float* Wa   = (const float*)d_in[8];   // [4,32,32]
  const float* ba   = (const float*)d_in[9];   // [4,32]
  const float* Wb   = (const float*)d_in[10];
  const float* bb   = (const float*)d_in[11];
  const float* gs   = (const float*)d_in[12];
  const float* bts  = (const float*)d_in[13];
  const float* fc_w = (const float*)d_in[14];  // [32,1024]
  const float* fc_b = (const float*)d_in[15];

  const int* srcI = ei;
  const int* dstI = ei + EE;

  float* ws    = (float*)d_ws;
  float* h128  = ws;                           // N*128 (layer-1 agg; reused)
  float* agg32 = ws;                           // alias: N*32 agg for layers 2-5
  float* xbuf  = ws + (size_t)NN * FIN;        // N*32 current features
  float* stats = xbuf + (size_t)NN * DIM;      // 64
  float* ss    = stats + 64;                   // 64 (scale/shift)
  v2f*   pw1a  = (v2f*)(ss + 64);              // 64*32   pairs
  v2f*   pw1b  = pw1a + (FIN / 2) * DIM;       // 16*32
  v2f*   pWa   = pw1b + (DIM / 2) * DIM;       // 4 * 16*32
  v2f*   pWb   = pWa + 4 * (DIM / 2) * DIM;    // 4 * 16*32
  v2f*   pfc   = pWb + 4 * (DIM / 2) * DIM;    // 16*1024

  const int T = 256;
  const unsigned scatterBlocks = EE / 8;       // one wave (32 lanes) per edge
  const int mlpBlocks = NN / (16 * 8);
  const int ewBlocks  = NN * DIM / 4 / T;      // float4 elementwise

  // ---- pack all weights into WMMA-B-pair layout ----
  k_packB<<<(FIN / 2 * DIM + T - 1) / T, T, 0, stream>>>(w1a, pw1a, FIN, DIM);
  k_packB<<<(DIM / 2 * DIM + T - 1) / T, T, 0, stream>>>(w1b, pw1b, DIM, DIM);
  for (int l = 0; l < 4; ++l) {
    k_packB<<<(DIM / 2 * DIM + T - 1) / T, T, 0, stream>>>(
        Wa + (size_t)l * DIM * DIM, pWa + (size_t)l * (DIM / 2) * DIM, DIM, DIM);
    k_packB<<<(DIM / 2 * DIM + T - 1) / T, T, 0, stream>>>(
        Wb + (size_t)l * DIM * DIM, pWb + (size_t)l * (DIM / 2) * DIM, DIM, DIM);
  }
  k_packB<<<(DIM / 2 * OUTD + T - 1) / T, T, 0, stream>>>(fc_w, pfc, DIM, OUTD);

  // ---- layer 1 (128 -> 32) ----
  k_copy4<<<NN * FIN / 4 / T, T, 0, stream>>>(
      (const float4*)node, (float4*)h128, NN * FIN / 4);
  k_scatter128<<<scatterBlocks, T, 0, stream>>>(node, srcI, dstI, h128);
  k_zero<<<1, 64, 0, stream>>>(stats, 64);
  k_mlp<FIN><<<mlpBlocks, T, 0, stream>>>(h128, pw1a, b1a, pw1b, b1b, xbuf, stats);
  k_bn_fin<<<1, 32, 0, stream>>>(stats, g1, bt1, ss);

  // ---- layers 2-5 (32 -> 32) ----
  for (int l = 0; l < 4; ++l) {
    // BN apply of previous layer fused with h = x copy for this layer
    k_bn_apply_dup<<<ewBlocks, T, 0, stream>>>(xbuf, agg32, ss);
    k_scatter32<<<scatterBlocks, T, 0, stream>>>(xbuf, srcI, dstI, agg32);
    k_zero<<<1, 64, 0, stream>>>(stats, 64);
    k_mlp<DIM><<<mlpBlocks, T, 0, stream>>>(
        agg32, pWa + (size_t)l * (DIM / 2) * DIM, ba + (size_t)l * DIM,
        pWb + (size_t)l * (DIM / 2) * DIM, bb + (size_t)l * DIM, xbuf, stats);
    k_bn_fin<<<1, 32, 0, stream>>>(stats, gs + (size_t)l * DIM, bts + (size_t)l * DIM, ss);
  }

  // ---- final BN apply, then FC (32 -> 1024) + relu ----
  k_bn_apply<<<ewBlocks, T, 0, stream>>>(xbuf, ss);
  k_fc<<<NN / 16, T, 0, stream>>>(xbuf, pfc, fc_b, (float*)d_out);
}